// SageLinkPrediction_54056458387940
// MI455X (gfx1250) — compile-verified
//
#include <hip/hip_runtime.h>
#include <hip/hip_bf16.h>

typedef __attribute__((ext_vector_type(2))) float v2f;
typedef __attribute__((ext_vector_type(8))) float v8f;

// ---------------------------------------------------------------- utilities

__global__ __launch_bounds__(256) void zero_kernel(float* __restrict__ p, size_t n) {
    size_t i = (size_t)blockIdx.x * blockDim.x + threadIdx.x;
    size_t stride = (size_t)gridDim.x * blockDim.x;
    for (; i < n; i += stride) p[i] = 0.0f;
}

__global__ __launch_bounds__(256) void degree_kernel(const int* __restrict__ dst,
                                                     float* __restrict__ deg, int E) {
    int e = blockIdx.x * blockDim.x + threadIdx.x;
    if (e < E) atomicAdd(&deg[dst[e]], 1.0f);
}

// one thread per (edge, 4-column chunk): float4 gather + 4 f32 atomics
__global__ __launch_bounds__(256) void scatter_kernel(const float* __restrict__ X,
                                                      const int* __restrict__ src,
                                                      const int* __restrict__ dst,
                                                      float* __restrict__ agg,
                                                      long long total, int chunks, int D) {
    long long tid = (long long)blockIdx.x * blockDim.x + threadIdx.x;
    if (tid >= total) return;
    int e = (int)(tid / chunks);
    int c = (int)(tid % chunks) * 4;
    int s = src[e], d = dst[e];
    float4 v = *(const float4*)(X + (size_t)s * D + c);
    float* a = agg + (size_t)d * D + c;
    atomicAdd(a + 0, v.x);
    atomicAdd(a + 1, v.y);
    atomicAdd(a + 2, v.z);
    atomicAdd(a + 3, v.w);
}

// ------------------------------------------------- fused SAGE layer (WMMA f32)
// Y[m, 0:256] = act( Xself[m,:] @ Wself + (AGG[m,:]/max(deg,1)) @ Wneigh + b )
// Block: 256 threads = 8 waves; 16 rows x 256 cols per block.
// Each wave: two 16x16 tiles (n0 = wave*16 and wave*16+128),
// K-loop in steps of 4 using V_WMMA_F32_16X16X4_F32.
template <int K>
__global__ __launch_bounds__(256) void sage_layer_kernel(
    const float* __restrict__ X,      // (>=n_dst, K) self features
    const float* __restrict__ AGG,    // (n_dst, K) summed neighbor features
    const float* __restrict__ DEG,    // (n_dst)
    const float* __restrict__ Wself,  // (K, 256) row-major
    const float* __restrict__ Wneigh, // (K, 256) row-major
    const float* __restrict__ bias,   // (256)
    float* __restrict__ Y,            // (n_dst, 256)
    int n_dst, int do_relu) {
    constexpr int N = 256;
    constexpr int LDK = K + 2;  // pad: lanes 0..15 read row*LDK -> distinct banks
    __shared__ __align__(16) float sSelf[16 * LDK];
    __shared__ __align__(16) float sMean[16 * LDK];
    __shared__ float sRdeg[16];

    const int tid = threadIdx.x;
    const int m0 = blockIdx.x * 16;

    if (tid < 16) {
        int row = m0 + tid;
        float d = (row < n_dst) ? DEG[row] : 1.0f;
        sRdeg[tid] = 1.0f / fmaxf(d, 1.0f);
    }
    __syncthreads();

    // stage 16 rows of self + mean features into LDS
    for (int idx = tid; idx < 16 * K; idx += 256) {
        int r = idx / K, c = idx - r * K;
        int row = m0 + r;
        float xs = 0.0f, xm = 0.0f;
        if (row < n_dst) {
            xs = X[(size_t)row * K + c];
            xm = AGG[(size_t)row * K + c] * sRdeg[r];
        }
        sSelf[r * LDK + c] = xs;
        sMean[r * LDK + c] = xm;
    }
    __syncthreads();

    const int wave = tid >> 5;
    const int lane = tid & 31;
    const int arow = lane & 15;          // A fragment: M index
    const int ksel = (lane >> 4) << 1;   // lanes 0-15: K=k,k+1 ; lanes 16-31: K=k+2,k+3
    const int ncol = lane & 15;          // B/C fragment: N index within tile
    const int n0a = wave * 16;
    const int n0b = n0a + 128;

    v8f acc0 = {};
    v8f acc1 = {};

    for (int k = 0; k < K; k += 4) {
        const int koff = k + ksel;
        v2f aS = *(const v2f*)&sSelf[arow * LDK + koff];
        v2f aM = *(const v2f*)&sMean[arow * LDK + koff];

        const float* ws = Wself + (size_t)koff * N;
        const float* wn = Wneigh + (size_t)koff * N;
        v2f bS0, bN0, bS1, bN1;
        bS0.x = ws[n0a + ncol];     bS0.y = ws[N + n0a + ncol];
        bN0.x = wn[n0a + ncol];     bN0.y = wn[N + n0a + ncol];
        bS1.x = ws[n0b + ncol];     bS1.y = ws[N + n0b + ncol];
        bN1.x = wn[n0b + ncol];     bN1.y = wn[N + n0b + ncol];

        acc0 = __builtin_amdgcn_wmma_f32_16x16x4_f32(false, aS, false, bS0, (short)0, acc0, false, false);
        acc0 = __builtin_amdgcn_wmma_f32_16x16x4_f32(false, aM, false, bN0, (short)0, acc0, false, false);
        acc1 = __builtin_amdgcn_wmma_f32_16x16x4_f32(false, aS, false, bS1, (short)0, acc1, false, false);
        acc1 = __builtin_amdgcn_wmma_f32_16x16x4_f32(false, aM, false, bN1, (short)0, acc1, false, false);
    }

    // C/D layout: VGPR r -> (M = r, lanes 0-15) / (M = r+8, lanes 16-31), N = lane&15
    const int mbase = m0 + ((lane >> 4) << 3);
    const float ba = bias[n0a + ncol];
    const float bb = bias[n0b + ncol];
#pragma unroll
    for (int r = 0; r < 8; ++r) {
        int m = mbase + r;
        if (m < n_dst) {
            float va = acc0[r] + ba;
            float vb = acc1[r] + bb;
            if (do_relu) {
                va = fmaxf(va, 0.0f);
                vb = fmaxf(vb, 0.0f);
            }
            Y[(size_t)m * N + n0a + ncol] = va;
            Y[(size_t)m * N + n0b + ncol] = vb;
        }
    }
}

// ---------------------------------------------------------------- decoder
// one wave32 per edge: logits[e] = dot(H[u], H[v]), N = 256
__global__ __launch_bounds__(256) void decode_kernel(const float* __restrict__ H,
                                                     const int* __restrict__ U,
                                                     const int* __restrict__ V,
                                                     float* __restrict__ out, int E, int N) {
    int warp = blockIdx.x * (blockDim.x >> 5) + (threadIdx.x >> 5);
    int lane = threadIdx.x & 31;
    if (warp >= E) return;
    const float* hu = H + (size_t)U[warp] * N;
    const float* hv = H + (size_t)V[warp] * N;
    float s = 0.0f;
    for (int c = lane * 4; c < N; c += 32 * 4) {
        float4 a = *(const float4*)(hu + c);
        float4 b = *(const float4*)(hv + c);
        s += a.x * b.x + a.y * b.y + a.z * b.z + a.w * b.w;
    }
#pragma unroll
    for (int off = 16; off > 0; off >>= 1) s += __shfl_xor(s, off, 32);
    if (lane == 0) out[warp] = s;
}

// ---------------------------------------------------------------- launch

extern "C" void kernel_launch(void* const* d_in, const int* in_sizes, int n_in,
                              void* d_out, int out_size, void* d_ws, size_t ws_size,
                              hipStream_t stream) {
    const float* X      = (const float*)d_in[0];
    const int*   src0   = (const int*)d_in[1];
    const int*   dst0   = (const int*)d_in[2];
    const int*   src1   = (const int*)d_in[3];
    const int*   dst1   = (const int*)d_in[4];
    const int*   decu   = (const int*)d_in[5];
    const int*   decv   = (const int*)d_in[6];
    const float* Wself0 = (const float*)d_in[8];
    const float* Wneigh0= (const float*)d_in[9];
    const float* b0     = (const float*)d_in[10];
    const float* Wself1 = (const float*)d_in[11];
    const float* Wneigh1= (const float*)d_in[12];
    const float* b1     = (const float*)d_in[13];

    const int E0 = in_sizes[1], E1 = in_sizes[3], EDEC = in_sizes[5];
    constexpr int NDST0 = 100000, NDST1 = 20000;
    constexpr int DIN = 128, DHID = 256, DOUT = 256;

    float* ws   = (float*)d_ws;
    float* agg0 = ws;
    float* deg0 = agg0 + (size_t)NDST0 * DIN;
    float* h0   = deg0 + NDST0;
    float* agg1 = h0 + (size_t)NDST0 * DHID;
    float* deg1 = agg1 + (size_t)NDST1 * DHID;
    float* h1   = deg1 + NDST1;

    // ---- layer 0 ----
    size_t z0 = (size_t)NDST0 * DIN + NDST0;  // agg0 + deg0 are contiguous
    zero_kernel<<<4096, 256, 0, stream>>>(agg0, z0);
    degree_kernel<<<(E0 + 255) / 256, 256, 0, stream>>>(dst0, deg0, E0);
    {
        long long total = (long long)E0 * (DIN / 4);
        int blocks = (int)((total + 255) / 256);
        scatter_kernel<<<blocks, 256, 0, stream>>>(X, src0, dst0, agg0, total, DIN / 4, DIN);
    }
    sage_layer_kernel<DIN><<<NDST0 / 16, 256, 0, stream>>>(X, agg0, deg0, Wself0, Wneigh0, b0,
                                                           h0, NDST0, /*relu=*/1);

    // ---- layer 1 ----
    size_t z1 = (size_t)NDST1 * DHID + NDST1;  // agg1 + deg1 contiguous
    zero_kernel<<<2048, 256, 0, stream>>>(agg1, z1);
    degree_kernel<<<(E1 + 255) / 256, 256, 0, stream>>>(dst1, deg1, E1);
    {
        long long total = (long long)E1 * (DHID / 4);
        int blocks = (int)((total + 255) / 256);
        scatter_kernel<<<blocks, 256, 0, stream>>>(h0, src1, dst1, agg1, total, DHID / 4, DHID);
    }
    sage_layer_kernel<DHID><<<NDST1 / 16, 256, 0, stream>>>(h0, agg1, deg1, Wself1, Wneigh1, b1,
                                                            h1, NDST1, /*relu=*/0);

    // ---- decoder ----
    decode_kernel<<<(EDEC + 7) / 8, 256, 0, stream>>>(h1, decu, decv, (float*)d_out, EDEC, DOUT);
}